// QuantumQuanvolutionFilter_65481071399473
// MI455X (gfx1250) — compile-verified
//
#include <hip/hip_runtime.h>

// ---------------------------------------------------------------------------
// out[B,4] = probs[B,16] @ S[16,4], S[s][n] = (-1)^{b_n(s)} (CNOT ring folded
// into the sign matrix). probs factorize per qubit, so each lane builds only
// the 8 A-matrix entries it owns. Contraction: 4 accumulating
// V_WMMA_F32_16X16X4_F32 per 16-element batch tile (exact f32 math).
// D is transposed through LDS (stride-20 rows, bank-conflict-free) so the
// write stream is one coalesced 256B global_store_b64 per tile.
// B sign matrix is pinned in VGPRs via empty inline asm so the backend
// cannot rematerialize it inside the tile loop.
// Memory-bound: 64 MiB total traffic -> ~2.8 us at 23.3 TB/s.
// ---------------------------------------------------------------------------

typedef float v2f __attribute__((ext_vector_type(2)));
typedef float v4f __attribute__((ext_vector_type(4)));
typedef float v8f __attribute__((ext_vector_type(8)));

#define TILES_PER_WAVE 4
#define LDS_ROW 20                  // floats per row: 80B, conflict-free & aligned

__global__ __launch_bounds__(256) void qexp_wmma_kernel(
    const float* __restrict__ x, float* __restrict__ out,
    int B, int ntiles) {
  __shared__ float lds[8 * 16 * LDS_ROW];          // 8 waves * 16x20 f32

  const int lane = threadIdx.x & 31;
  const int n    = lane & 15;   // WMMA column index (only 0..3 live)
  const int half = lane >> 4;   // selects K rows {2h, 2h+1} of each chunk
  const int wib  = threadIdx.x >> 5;
  const int waveId = blockIdx.x * 8 + wib;
  const int ldsBase = wib * (16 * LDS_ROW);

  // ---- lane-constant B matrix (sign matrix), computed once per wave ----
  // chunk g, reg r holds S'[s][n] with s = 4g + 2*half + r,
  // bits: i=(g>>1)&1, j=g&1, k=half, l=r. Columns n>=4 are zero padding.
  const float colMask = (n < 4) ? 1.0f : 0.0f;
  float bm[4][2];
#pragma unroll
  for (int g = 0; g < 4; ++g) {
#pragma unroll
    for (int r = 0; r < 2; ++r) {
      const int i = (g >> 1) & 1, j = g & 1, k = half, l = r;
      int bit;
      if      (n == 0) bit = j ^ k ^ l;
      else if (n == 1) bit = i ^ j;
      else if (n == 2) bit = i ^ j ^ k;
      else             bit = i ^ j ^ k ^ l;   // n == 3 and don't-cares
      bm[g][r] = colMask * (bit ? -1.0f : 1.0f);
    }
  }
  v2f b0v = { bm[0][0], bm[0][1] };
  v2f b1v = { bm[1][0], bm[1][1] };
  v2f b2v = { bm[2][0], bm[2][1] };
  v2f b3v = { bm[3][0], bm[3][1] };
  // Pin the sign matrix in VGPRs: opaque to the optimizer, so the cndmask
  // chains cannot be rematerialized inside the tile loop.
  asm("" : "+v"(b0v), "+v"(b1v), "+v"(b2v), "+v"(b3v));

  const float sgn2 = half ? -0.5f : 0.5f;   // p2(k = half) = 0.5 + sgn2*cos

  // readback mapping: lane -> (element, which 8B half of its 4-float row)
  const int elem    = lane >> 1;
  const int pairsel = lane & 1;

  const int tile0 = waveId * TILES_PER_WAVE;
#pragma unroll
  for (int t = 0; t < TILES_PER_WAVE; ++t) {
    const int tile = tile0 + t;
    if (tile >= ntiles) break;              // wave-uniform branch
    const int mBase = tile << 4;

    // One float4 per lane (lanes 16..31 mirror 0..15 -> cache hit).
    // Clamp so EXEC stays all-ones through the WMMAs.
    int m  = mBase + n;
    int mc = m < B ? m : B - 1;
    const v4f th = __builtin_nontemporal_load(
        reinterpret_cast<const v4f*>(x) + mc);

    const float c0 = __cosf(th.x), c1 = __cosf(th.y),
                c2 = __cosf(th.z), c3 = __cosf(th.w);

    // per-qubit probabilities (1 +/- cos)/2
    const float p0_0 = 0.5f + 0.5f * c0, p0_1 = 0.5f - 0.5f * c0;
    const float p1_0 = 0.5f + 0.5f * c1, p1_1 = 0.5f - 0.5f * c1;
    const float p2h  = 0.5f + sgn2 * c2;                 // p2(k = half)
    const float p3_0 = 0.5f + 0.5f * c3, p3_1 = 0.5f - 0.5f * c3;

    // pair products: q_g = p0(i)*p1(j) with g = 2i+j; rk_r = p2(half)*p3(r)
    const float q0 = p0_0 * p1_0, q1 = p0_0 * p1_1;
    const float q2 = p0_1 * p1_0, q3 = p0_1 * p1_1;
    const float rk0 = p2h * p3_0, rk1 = p2h * p3_1;

    // A-matrix chunks (16x4 f32, 2 VGPRs): reg r = prob(4g + 2*half + r)
    const v2f a0 = { q0 * rk0, q0 * rk1 };
    const v2f a1 = { q1 * rk0, q1 * rk1 };
    const v2f a2 = { q2 * rk0, q2 * rk1 };
    const v2f a3 = { q3 * rk0, q3 * rk1 };

    v8f d = {0.f, 0.f, 0.f, 0.f, 0.f, 0.f, 0.f, 0.f};
    d = __builtin_amdgcn_wmma_f32_16x16x4_f32(false, a0, false, b0v,
                                              (short)0, d, false, false);
    d = __builtin_amdgcn_wmma_f32_16x16x4_f32(false, a1, false, b1v,
                                              (short)0, d, false, false);
    d = __builtin_amdgcn_wmma_f32_16x16x4_f32(false, a2, false, b2v,
                                              (short)0, d, false, false);
    d = __builtin_amdgcn_wmma_f32_16x16x4_f32(false, a3, false, b3v,
                                              (short)0, d, false, false);

    // Scatter D into LDS in its natural layout: VGPR r holds
    // (M = r + 8*half, N = n). All 32 lanes write -> full 16x16 tile,
    // no predication, conflict-free with stride-20 rows.
#pragma unroll
    for (int r = 0; r < 8; ++r) {
      lds[ldsBase + (r + (half << 3)) * LDS_ROW + n] = d[r];
    }
    __builtin_amdgcn_wave_barrier();        // same-wave DS ops are in-order;
                                            // just block compiler reordering

    // Coalesced readback: lane L -> 8 bytes of element L/2's output row.
    const v2f v = *reinterpret_cast<const v2f*>(
        &lds[ldsBase + elem * LDS_ROW + pairsel * 2]);
    const int mg = mBase + elem;
    if (mg < B) {                           // single per-lane predicate
      __builtin_nontemporal_store(
          v, reinterpret_cast<v2f*>(out + mg * 4 + pairsel * 2));
    }
    __builtin_amdgcn_wave_barrier();        // keep next tile's DS writes after
  }
}

extern "C" void kernel_launch(void* const* d_in, const int* in_sizes, int n_in,
                              void* d_out, int out_size, void* d_ws, size_t ws_size,
                              hipStream_t stream) {
  const float* x = (const float*)d_in[0];
  float* out = (float*)d_out;
  const int B = in_sizes[0] / 4;                       // batch (x is [B,4])
  const int ntiles = (B + 15) / 16;
  int waves = (ntiles + TILES_PER_WAVE - 1) / TILES_PER_WAVE;
  int blocks = (waves + 7) / 8;                        // 8 wave32 per 256-thr block
  if (blocks < 1) blocks = 1;
  qexp_wmma_kernel<<<blocks, 256, 0, stream>>>(x, out, B, ntiles);
}